// MF_26199300506017
// MI455X (gfx1250) — compile-verified
//
#include <hip/hip_runtime.h>
#include <hip/hip_bf16.h>
#include <math.h>

// Cosine-similarity full matrix on MI455X (gfx1250, wave32, WMMA).
//   C[u,i] = (U_u . V_i) * invnorm(U_u) * invnorm(V_i)
// M = N = 8192, K = 256 (f32). Output-store bound (256 MB @ 23.3 TB/s ~ 11us).
// f32 WMMA path (V_WMMA_F32_16X16X4_F32), LDS double-buffered 128x128 tile.
// Global->LDS staging uses GLOBAL_LOAD_ASYNC_TO_LDS_B128 (ASYNCcnt) so the
// prefetch holds no VGPRs (fixes the scratch-spill + forced-wait serialization
// seen with register staging). Fragments are single conflict-free ds_load_b64.

typedef __attribute__((ext_vector_type(2))) float v2f;
typedef __attribute__((ext_vector_type(8))) float v8f;

#define L_DIM 256            // embedding length (K)
#define BM 128               // block tile M
#define BN 128               // block tile N
#define KC 16                // K chunk staged in LDS
#define LDS_K (KC + 4)       // row stride 20 floats = 80 B (16B-aligned rows)

#if defined(__has_builtin)
#if __has_builtin(__builtin_amdgcn_global_load_async_to_lds_b128)
#define HAVE_ASYNC_COPY 1
#endif
#endif
#ifndef HAVE_ASYNC_COPY
#define HAVE_ASYNC_COPY 0
#endif

// Builtin signature (from hipcc diagnostic): src/dst are int4 pointers in
// AS1 (global) / AS3 (LDS), plus imm offset and imm cpol.
typedef int v4i_t __attribute__((vector_size(16)));
typedef __attribute__((address_space(1))) v4i_t* gvec4_t;
typedef __attribute__((address_space(3))) v4i_t* lvec4_t;

__device__ __forceinline__ void wait_async_zero() {
#if defined(__has_builtin) && __has_builtin(__builtin_amdgcn_s_wait_asynccnt)
  __builtin_amdgcn_s_wait_asynccnt(0);
#else
  asm volatile("s_wait_asynccnt 0x0" ::: "memory");
#endif
}

// ---------------------------------------------------------------------------
// Row inverse-norm kernel: one wave32 per row.
// ---------------------------------------------------------------------------
__global__ __launch_bounds__(256) void cos_rownorm_kernel(
    const float* __restrict__ W, float* __restrict__ inv_out, int nrows) {
  const int lane = threadIdx.x & 31;
  const int row = blockIdx.x * 8 + (threadIdx.x >> 5);
  if (row >= nrows) return;
  const float* p = W + (size_t)row * L_DIM;
  float s = 0.0f;
#pragma unroll
  for (int k = 0; k < L_DIM / 32; ++k) {
    float x = p[lane + k * 32];
    s = fmaf(x, x, s);
  }
#pragma unroll
  for (int off = 16; off > 0; off >>= 1) s += __shfl_xor(s, off, 32);
  if (lane == 0) inv_out[row] = 1.0f / fmaxf(sqrtf(s), 1e-8f);
}

// ---------------------------------------------------------------------------
// WMMA GEMM with normalization epilogue.
// Block: 256 threads = 8 waves arranged 2 (M) x 4 (N).
// Wave tile: 64 x 32 = 4x2 grid of 16x16 f32 WMMA accumulators.
// ---------------------------------------------------------------------------
__global__ __launch_bounds__(256)
__attribute__((amdgpu_waves_per_eu(1))) void cos_gemm_kernel(
    const float* __restrict__ A,     // user embeddings [M][256]
    const float* __restrict__ B,     // item embeddings [N][256]
    const float* __restrict__ invU,  // [M]
    const float* __restrict__ invV,  // [N]
    float* __restrict__ out,         // [M][N]
    int Ncols) {
  __shared__ __align__(16) float As[2][BM][LDS_K];  // M-major A tile (m, k)
  __shared__ __align__(16) float Bs[2][BN][LDS_K];  // N-major B tile (n, k)

  const int tid = threadIdx.x;
  const int lane = tid & 31;
  const int half = lane >> 4;  // 0: lanes 0-15 (k0,k0+1), 1: lanes 16-31 (+2,+3)
  const int l = lane & 15;
  const int wave = tid >> 5;    // 0..7
  const int waveM = wave >> 2;  // 0..1
  const int waveN = wave & 3;   // 0..3

  const int blockM = blockIdx.y * BM;
  const int blockN = blockIdx.x * BN;
  const int m_w = waveM * 64;
  const int n_w = waveN * 32;

  v8f acc[4][2] = {};  // zero-initialized accumulators

  auto compute = [&](int buf) {
#pragma unroll
    for (int ks = 0; ks < KC; ks += 4) {
      const int kf = ks + 2 * half;  // ISA 16x4 f32 fragment layout
      v2f bfrag[2];
#pragma unroll
      for (int tj = 0; tj < 2; ++tj)
        bfrag[tj] = *(const v2f*)&Bs[buf][n_w + tj * 16 + l][kf];
      v2f afrag[4];
#pragma unroll
      for (int ti = 0; ti < 4; ++ti)
        afrag[ti] = *(const v2f*)&As[buf][m_w + ti * 16 + l][kf];
#pragma unroll
      for (int ti = 0; ti < 4; ++ti) {
#pragma unroll
        for (int tj = 0; tj < 2; ++tj) {
          acc[ti][tj] = __builtin_amdgcn_wmma_f32_16x16x4_f32(
              /*neg_a=*/false, afrag[ti], /*neg_b=*/false, bfrag[tj],
              /*c_mod=*/(short)0, acc[ti][tj],
              /*reuse_a=*/false, /*reuse_b=*/false);
        }
      }
    }
  };

#if HAVE_ASYNC_COPY
  // -------- Async global->LDS double buffering (no staging VGPRs). --------
  auto issue_chunk = [&](int kc, int buf) {
#pragma unroll
    for (int i = 0; i < 2; ++i) {
      const int q = tid + i * 256;  // 0..511
      const int row = q >> 2;       // 0..127
      const int kq = (q & 3) << 2;  // 0,4,8,12
      __builtin_amdgcn_global_load_async_to_lds_b128(
          (gvec4_t)(A + (size_t)(blockM + row) * L_DIM + kc + kq),
          (lvec4_t)&As[buf][row][kq], 0, 0);
      __builtin_amdgcn_global_load_async_to_lds_b128(
          (gvec4_t)(B + (size_t)(blockN + row) * L_DIM + kc + kq),
          (lvec4_t)&Bs[buf][row][kq], 0, 0);
    }
  };

  issue_chunk(0, 0);
  wait_async_zero();
  __syncthreads();

  int buf = 0;
  for (int kc = 0; kc < L_DIM; kc += KC) {
    const bool has_next = (kc + KC) < L_DIM;
    if (has_next) issue_chunk(kc + KC, buf ^ 1);  // in flight during compute
    compute(buf);
    if (has_next) wait_async_zero();  // my copies into buf^1 landed
    __syncthreads();                  // everyone's copies visible
    buf ^= 1;
  }
#else
  // -------- Fallback: register-staged double buffering. --------
  float4 ra[2], rb[2];
  auto fetch = [&](int kc) {
#pragma unroll
    for (int i = 0; i < 2; ++i) {
      const int q = tid + i * 256;
      const int row = q >> 2;
      const int kq = (q & 3) << 2;
      ra[i] = *(const float4*)(A + (size_t)(blockM + row) * L_DIM + kc + kq);
      rb[i] = *(const float4*)(B + (size_t)(blockN + row) * L_DIM + kc + kq);
    }
  };
  auto stage = [&](int buf) {
#pragma unroll
    for (int i = 0; i < 2; ++i) {
      const int q = tid + i * 256;
      const int row = q >> 2;
      const int kq = (q & 3) << 2;
      *(float4*)&As[buf][row][kq] = ra[i];
      *(float4*)&Bs[buf][row][kq] = rb[i];
    }
  };

  fetch(0);
  stage(0);
  __syncthreads();

  int buf = 0;
  for (int kc = 0; kc < L_DIM; kc += KC) {
    const bool has_next = (kc + KC) < L_DIM;
    if (has_next) fetch(kc + KC);
    compute(buf);
    if (has_next) stage(buf ^ 1);
    __syncthreads();
    buf ^= 1;
  }
#endif

  // Epilogue: scale by inverse norms; nontemporal (streaming) stores since
  // the 256MB result is write-once and must not evict A/B tiles from L2.
  const int mBase = blockM + m_w;
  const int nBase = blockN + n_w;
#pragma unroll
  for (int ti = 0; ti < 4; ++ti) {
#pragma unroll
    for (int tj = 0; tj < 2; ++tj) {
      const int n = nBase + tj * 16 + l;
      const float sv = invV[n];
#pragma unroll
      for (int r = 0; r < 8; ++r) {
        const int m = mBase + ti * 16 + half * 8 + r;
        __builtin_nontemporal_store(acc[ti][tj][r] * invU[m] * sv,
                                    out + (size_t)m * (size_t)Ncols + n);
      }
    }
  }
}

// ---------------------------------------------------------------------------
// Launch: inputs are f32 user_embed_w [U*256], item_embed_w [I*256],
// user_idx, item_idx, is_test (test path = full cosine matrix).
// ---------------------------------------------------------------------------
extern "C" void kernel_launch(void* const* d_in, const int* in_sizes, int n_in,
                              void* d_out, int out_size, void* d_ws,
                              size_t ws_size, hipStream_t stream) {
  const float* Uw = (const float*)d_in[0];
  const float* Vw = (const float*)d_in[1];
  const int U = in_sizes[0] / L_DIM;  // 8192
  const int I = in_sizes[1] / L_DIM;  // 8192

  float* invU = (float*)d_ws;
  float* invV = invU + U;

  cos_rownorm_kernel<<<(U + 7) / 8, 256, 0, stream>>>(Uw, invU, U);
  cos_rownorm_kernel<<<(I + 7) / 8, 256, 0, stream>>>(Vw, invV, I);

  dim3 grid(I / BN, U / BM);  // 64 x 64 blocks
  cos_gemm_kernel<<<grid, 256, 0, stream>>>(Uw, Vw, invU, invV, (float*)d_out,
                                            I);
}